// DFTBSK_44676249813578
// MI455X (gfx1250) — compile-verified
//
#include <hip/hip_runtime.h>
#include <hip/hip_bf16.h>
#include <stdint.h>

// Problem constants (from the reference)
#define RSZ 13            // Slater-Koster rows per bond type
#define GSZ 512           // distance grid points
#define OE_SZ 4           // onsite entries per atom
#define TPB 256           // threads per block (8 wave32s)
#define STAGE_STRIDE 257  // pad LDS rows to de-conflict 64 banks

// Native clang vector: required by __builtin_nontemporal_store/load,
// lowers to b128 global memory ops.
typedef float v4f __attribute__((ext_vector_type(4)));

// ---------------------------------------------------------------------------
// Edge kernel: fused gather + linear interpolation for hopping & overlap.
// One edge per lane; results staged in LDS and flushed with coalesced
// non-temporal v4f (b128) stores. xx grid staged via CDNA5 async
// global->LDS loads (ASYNCcnt). Streaming inputs use NT load hints so the
// L2 stays dedicated to the 5.3MB interpolation tables.
// ---------------------------------------------------------------------------
__global__ __launch_bounds__(TPB) void dftbsk_edge_kernel(
    const float* __restrict__ rij,
    const int*   __restrict__ edge_type,
    const float* __restrict__ xx,
    const float* __restrict__ hopping,   // [B][13][512]
    const float* __restrict__ overlap,   // [B][13][512]
    float* __restrict__ out_h,           // [E][13]
    float* __restrict__ out_o,           // [E][13]
    int E, int G)
{
    __shared__ float s_xx[GSZ];
    __shared__ float s_h[RSZ * STAGE_STRIDE];
    __shared__ float s_o[RSZ * STAGE_STRIDE];

    const int tid = threadIdx.x;

    // --- Stage the distance grid into LDS via the gfx1250 async-to-LDS path.
    // Generic address of a __shared__ object: low 32 bits are the LDS byte
    // address (ISA 10.2); that value goes in VDST.
    for (int i = tid; i < G; i += TPB) {
        uint32_t lds_addr = (uint32_t)(uintptr_t)(&s_xx[i]);
        const float* gptr = xx + i;
        asm volatile("global_load_async_to_lds_b32 %0, %1, off"
                     :: "v"(lds_addr), "v"(gptr) : "memory");
    }
    asm volatile("s_wait_asynccnt 0" ::: "memory");
    __syncthreads();

    const long long blockStart = (long long)blockIdx.x * TPB;
    long long remain = (long long)E - blockStart;
    const int count = remain > TPB ? TPB : (int)remain;   // edges in this block
    const long long e = blockStart + tid;

    if (tid < count) {
        // Streaming, never-reused inputs: non-temporal loads.
        const float r  = __builtin_nontemporal_load(rij + e);
        const int   et = __builtin_nontemporal_load(edge_type + e);

        // xx = linspace(1.0, 6.0, G): closed-form interval index, clamped.
        const float invdx = (float)(G - 1) * (1.0f / 5.0f);
        int idx = (int)((r - 1.0f) * invdx);
        idx = max(0, min(G - 2, idx));

        const float x0 = s_xx[idx];
        const float x1 = s_xx[idx + 1];
        const float t  = (r - x0) / (x1 - x0);

        const long long rowbase = ((long long)et * RSZ) * GSZ + idx;
        const float* hb = hopping + rowbase;
        const float* ob = overlap + rowbase;

#pragma unroll
        for (int rr = 0; rr < RSZ; ++rr) {
            // Adjacent grid points: L2-resident gathers with default RT hint
            // (tables are 5.3MB total; NT output stores keep them cached).
            const float h0 = hb[rr * GSZ];
            const float h1 = hb[rr * GSZ + 1];
            const float o0 = ob[rr * GSZ];
            const float o1 = ob[rr * GSZ + 1];
            // y0*(1-t) + y1*t  ==  y0 + t*(y1-y0)  -> single v_fma_f32
            s_h[rr * STAGE_STRIDE + tid] = fmaf(t, h1 - h0, h0);
            s_o[rr * STAGE_STRIDE + tid] = fmaf(t, o1 - o0, o0);
        }
    }
    __syncthreads();

    // --- Coalesced flush: count*13 contiguous floats per output stream.
    // Vectorized b128 non-temporal stores (base byte offset = blockStart*52,
    // 16B-aligned since blockStart is a multiple of 256). Outputs are
    // write-once -> TH_STORE_NT avoids evicting the SK tables from L2.
    const long long base  = blockStart * RSZ;   // float index of block's slab
    const int total  = count * RSZ;
    const int total4 = total >> 2;

    v4f* __restrict__ out_h4 = (v4f*)(out_h + base);
    v4f* __restrict__ out_o4 = (v4f*)(out_o + base);

    for (int k4 = tid; k4 < total4; k4 += TPB) {
        const int k = k4 << 2;
        v4f vh, vo;
        {
            int er = (k + 0) / RSZ, rr = (k + 0) - er * RSZ;
            vh.x = s_h[rr * STAGE_STRIDE + er];
            vo.x = s_o[rr * STAGE_STRIDE + er];
        }
        {
            int er = (k + 1) / RSZ, rr = (k + 1) - er * RSZ;
            vh.y = s_h[rr * STAGE_STRIDE + er];
            vo.y = s_o[rr * STAGE_STRIDE + er];
        }
        {
            int er = (k + 2) / RSZ, rr = (k + 2) - er * RSZ;
            vh.z = s_h[rr * STAGE_STRIDE + er];
            vo.z = s_o[rr * STAGE_STRIDE + er];
        }
        {
            int er = (k + 3) / RSZ, rr = (k + 3) - er * RSZ;
            vh.w = s_h[rr * STAGE_STRIDE + er];
            vo.w = s_o[rr * STAGE_STRIDE + er];
        }
        __builtin_nontemporal_store(vh, out_h4 + k4);
        __builtin_nontemporal_store(vo, out_o4 + k4);
    }
    // Scalar tail (when count*13 isn't a multiple of 4).
    for (int k = (total4 << 2) + tid; k < total; k += TPB) {
        int er = k / RSZ, rr = k - er * RSZ;
        __builtin_nontemporal_store(s_h[rr * STAGE_STRIDE + er], out_h + base + k);
        __builtin_nontemporal_store(s_o[rr * STAGE_STRIDE + er], out_o + base + k);
    }
}

// ---------------------------------------------------------------------------
// Node kernel: node_features[n][0:4] = onsite_param[atom_type[n]][0:4].
// 160B onsite table stays resident; b128 gather + b128 coalesced NT store.
// ---------------------------------------------------------------------------
__global__ __launch_bounds__(TPB) void dftbsk_node_kernel(
    const int*   __restrict__ atom_type,
    const float* __restrict__ onsite,    // [T][4]
    float* __restrict__ out,             // [N][4]
    int N)
{
    const int n = blockIdx.x * TPB + threadIdx.x;
    if (n < N) {
        const int at = __builtin_nontemporal_load(atom_type + n);
        const v4f v = *(const v4f*)(onsite + (long long)at * OE_SZ);
        __builtin_nontemporal_store(v, (v4f*)(out + (long long)n * OE_SZ));
    }
}

extern "C" void kernel_launch(void* const* d_in, const int* in_sizes, int n_in,
                              void* d_out, int out_size, void* d_ws, size_t ws_size,
                              hipStream_t stream) {
    // setup_inputs() order:
    //   0 rij [E] f32, 1 edge_type [E] i32, 2 atom_type [N] i32,
    //   3 xx [G] f32, 4 hopping_param [B,13,G] f32,
    //   5 overlap_param [B,13,G] f32, 6 onsite_param [T,4,1] f32
    const float* rij       = (const float*)d_in[0];
    const int*   edge_type = (const int*)  d_in[1];
    const int*   atom_type = (const int*)  d_in[2];
    const float* xx        = (const float*)d_in[3];
    const float* hopping   = (const float*)d_in[4];
    const float* overlap   = (const float*)d_in[5];
    const float* onsite    = (const float*)d_in[6];

    const int E = in_sizes[0];
    const int N = in_sizes[2];
    const int G = in_sizes[3];

    // d_out = [edge_features E*13 | edge_overlap E*13 | node_features N*4]
    float* out   = (float*)d_out;
    float* out_h = out;
    float* out_o = out + (long long)E * RSZ;
    float* out_n = out + 2LL * (long long)E * RSZ;

    const int eblocks = (E + TPB - 1) / TPB;
    dftbsk_edge_kernel<<<eblocks, TPB, 0, stream>>>(
        rij, edge_type, xx, hopping, overlap, out_h, out_o, E, G);

    const int nblocks = (N + TPB - 1) / TPB;
    dftbsk_node_kernel<<<nblocks, TPB, 0, stream>>>(atom_type, onsite, out_n, N);
}